// Quaternion_Multiplicative_ErrorV2_74294344286531
// MI455X (gfx1250) — compile-verified
//
#include <hip/hip_runtime.h>

typedef float v2f __attribute__((ext_vector_type(2)));
typedef float v4f __attribute__((ext_vector_type(4)));
typedef float v8f __attribute__((ext_vector_type(8)));

// Sum a float across all 32 lanes of a wave.
// Uses V_WMMA_F32_16X16X4_F32: A = per-lane acc placed at (m, k=0) for lanes
// 0-15 and (m, k=2) for lanes 16-31 (per the 16x4 f32 A layout), B = ones.
// D[m][n] = acc_m + acc_{m+16}. Each lane then holds 8 of the 16 row-sums in
// its D VGPRs (lanes 0-15: rows 0-7, lanes 16-31: rows 8-15); summing those 8
// and one shfl_xor(16) yields the full 32-lane total in every lane.
// Must be called with EXEC all-ones (uniform control flow, full waves).
__device__ __forceinline__ float wave_sum32(float acc) {
#if __has_builtin(__builtin_amdgcn_wmma_f32_16x16x4_f32)
    v2f a; a[0] = acc;  a[1] = 0.0f;
    v2f b; b[0] = 1.0f; b[1] = 1.0f;
    v8f c = {};
    v8f d = __builtin_amdgcn_wmma_f32_16x16x4_f32(
        /*neg_a=*/false, a, /*neg_b=*/false, b,
        /*c_mod=*/(short)0, c, /*reuse_a=*/false, /*reuse_b=*/false);
    float r = (d[0] + d[1]) + (d[2] + d[3]);
    r += (d[4] + d[5]) + (d[6] + d[7]);
    r += __shfl_xor(r, 16, 32);
    return r;
#else
    for (int off = 16; off > 0; off >>= 1) acc += __shfl_xor(acc, off, 32);
    return acc;
#endif
}

// Stage 1: grid-stride streaming pass. One float4 (b128) nontemporal load per
// array per sample; per-thread accumulate |p1*t1|+|p2*t2|+|p3*t3|; then
// wave (WMMA) + LDS block reduction; one partial per block into d_ws.
__global__ void qme_partial_kernel(const float* __restrict__ pred,
                                   const float* __restrict__ tru,
                                   float* __restrict__ partial, int n) {
    const v4f* __restrict__ p4 = reinterpret_cast<const v4f*>(pred);
    const v4f* __restrict__ t4 = reinterpret_cast<const v4f*>(tru);

    float acc = 0.0f;
    const int stride = gridDim.x * blockDim.x;
    for (int i = blockIdx.x * blockDim.x + threadIdx.x; i < n; i += stride) {
        v4f p = __builtin_nontemporal_load(p4 + i);
        v4f t = __builtin_nontemporal_load(t4 + i);
        acc += __builtin_fabsf(p[1] * t[1]);
        acc += __builtin_fabsf(p[2] * t[2]);
        acc += __builtin_fabsf(p[3] * t[3]);
    }

    // Wave-level reduction (EXEC all-ones here: outside divergent loop).
    float wsum = wave_sum32(acc);

    __shared__ float smem[8];  // 256 threads -> 8 wave32s
    const int lane = threadIdx.x & 31;
    const int wid  = threadIdx.x >> 5;
    if (lane == 0) smem[wid] = wsum;
    __syncthreads();
    if (threadIdx.x == 0) {
        float bsum = 0.0f;
        const int nwaves = blockDim.x >> 5;
        for (int i = 0; i < nwaves; ++i) bsum += smem[i];
        partial[blockIdx.x] = bsum;
    }
}

// Stage 2: single block reduces the per-block partials, applies 2/B, writes
// the scalar mean. Fully deterministic (no atomics).
__global__ void qme_final_kernel(const float* __restrict__ partial, int nparts,
                                 float* __restrict__ out, float scale) {
    float acc = 0.0f;
    for (int i = threadIdx.x; i < nparts; i += blockDim.x) acc += partial[i];

    float wsum = wave_sum32(acc);

    __shared__ float smem[8];
    const int lane = threadIdx.x & 31;
    const int wid  = threadIdx.x >> 5;
    if (lane == 0) smem[wid] = wsum;
    __syncthreads();
    if (threadIdx.x == 0) {
        float total = 0.0f;
        const int nwaves = blockDim.x >> 5;
        for (int i = 0; i < nwaves; ++i) total += smem[i];
        out[0] = total * scale;
    }
}

extern "C" void kernel_launch(void* const* d_in, const int* in_sizes, int n_in,
                              void* d_out, int out_size, void* d_ws, size_t ws_size,
                              hipStream_t stream) {
    const float* pred = (const float*)d_in[0];
    const float* tru  = (const float*)d_in[1];
    float* out = (float*)d_out;

    const int n = in_sizes[0] / 4;  // number of quaternions (B)

    const int threads = 256;        // 8 wave32s per block
    int blocks = 4096;              // 32K waves in flight; 16 KiB of partials
    const size_t need = (size_t)blocks * sizeof(float);
    if (need > ws_size) blocks = (int)(ws_size / sizeof(float));
    if (blocks < 1) blocks = 1;

    float* partial = (float*)d_ws;

    qme_partial_kernel<<<blocks, threads, 0, stream>>>(pred, tru, partial, n);
    // 2/B scaling folded into the final write (2/8388608 is a power of two,
    // exact in fp32).
    qme_final_kernel<<<1, threads, 0, stream>>>(partial, blocks, out,
                                                2.0f / (float)n);
}